// FGLEncoder0_22411139350997
// MI455X (gfx1250) — compile-verified
//
#include <hip/hip_runtime.h>
#include <hip/hip_bf16.h>

typedef __attribute__((ext_vector_type(2))) float v2f;
typedef __attribute__((ext_vector_type(8))) float v8f;

#define N_BATCH 256
#define NIN     65536
#define J0      512
#define J1      32
#define C1      32
#define C2      128
#define E0CNT   (2 * NIN)   // 131072
#define E1CNT   (2 * 512)   // 1024
#define OUT_DIM 768

__device__ __forceinline__ int lower_bound_i32(const int* a, int n, int v) {
    int lo = 0, hi = n;
    while (lo < hi) { int m = (lo + hi) >> 1; if (a[m] < v) lo = m + 1; else hi = m; }
    return lo;
}

// D = A(16x4, f32) * B(4x16, f32) + C(16x16, f32)
__device__ __forceinline__ v8f wmma_f32(v2f a, v2f b, v8f c) {
    return __builtin_amdgcn_wmma_f32_16x16x4_f32(false, a, false, b, (short)0, c, false, false);
}

// ---------- x (256 x 65536) -> xT (65536 x 256), 32x32 LDS tiles ----------
__global__ void k_transpose_x(const float* __restrict__ x, float* __restrict__ xT) {
    __shared__ float tile[32][33];
    int c0 = blockIdx.x * 32;   // column in x
    int r0 = blockIdx.y * 32;   // row in x
    int tx = threadIdx.x, ty = threadIdx.y;
#pragma unroll
    for (int j = 0; j < 32; j += 8)
        tile[ty + j][tx] = x[(size_t)(r0 + ty + j) * NIN + (c0 + tx)];
    __syncthreads();
#pragma unroll
    for (int j = 0; j < 32; j += 8)
        xT[(size_t)(c0 + ty + j) * N_BATCH + (r0 + tx)] = tile[tx][ty + j];
}

// ---------- layer-0 scatter: h0T[j][n] = sum_{dst0[e]==j} x[n, src0[e]] ----------
__global__ void k_fgl0_gather(const float* __restrict__ xsrc, const int* __restrict__ src0,
                              const int* __restrict__ dst0, float* __restrict__ h0T,
                              int transposed) {
    int j = blockIdx.x;      // 0..511
    int n = threadIdx.x;     // 0..255
    int lo = lower_bound_i32(dst0, E0CNT, j);
    int hi = lower_bound_i32(dst0, E0CNT, j + 1);
    float acc = 0.f;
    if (transposed) {
        for (int e = lo; e < hi; ++e) {
            if (e + 32 < hi)  // hide L2 latency on the column gather
                __builtin_prefetch(xsrc + (size_t)src0[e + 32] * N_BATCH + n, 0, 0);
            acc += xsrc[(size_t)src0[e] * N_BATCH + n];  // coalesced 1KB column reads
        }
    } else {
        for (int e = lo; e < hi; ++e) acc += xsrc[(size_t)n * NIN + src0[e]];
    }
    h0T[j * N_BATCH + n] = acc;
}

// ---------- W0n[j][c] = g0[j]*V0[j,c,0]*rsqrt(sum_c V0^2) ----------
__global__ void k_w0norm(const float* __restrict__ V0, const float* __restrict__ g0,
                         float* __restrict__ W0n) {
    int j = blockIdx.x, c = threadIdx.x;  // 512 x 32
    float v = V0[j * C1 + c];
    float s = v * v;
#pragma unroll
    for (int off = 16; off > 0; off >>= 1) s += __shfl_xor(s, off, 32);
    W0n[j * C1 + c] = g0[j] * v * rsqrtf(s);
}

// ---------- W1n[j][o][c] = g1[j]*V1[j,o,c]*rsqrt(sum_{o,c} V1^2) ----------
__global__ void k_w1norm(const float* __restrict__ V1, const float* __restrict__ g1,
                         float* __restrict__ W1n) {
    __shared__ float red[256];
    int j = blockIdx.x, t = threadIdx.x;  // 32 x 256
    const float* Vj = V1 + j * (C2 * C1);
    float s = 0.f;
    for (int i = t; i < C2 * C1; i += 256) { float v = Vj[i]; s += v * v; }
    red[t] = s; __syncthreads();
    for (int w = 128; w > 0; w >>= 1) { if (t < w) red[t] += red[t + w]; __syncthreads(); }
    float rs = rsqrtf(red[0]) * g1[j];
    for (int i = t; i < C2 * C1; i += 256) W1n[j * (C2 * C1) + i] = Vj[i] * rs;
}

// ---------- Bs[jj][c] = sum_{dst1[e]==jj} b0[c, src1[e]] ----------
__global__ void k_bias0_scatter(const int* __restrict__ src1, const int* __restrict__ dst1,
                                const float* __restrict__ b0, float* __restrict__ Bs) {
    int jj = blockIdx.x, c = threadIdx.x;  // 32 x 32
    int lo = lower_bound_i32(dst1, E1CNT, jj);
    int hi = lower_bound_i32(dst1, E1CNT, jj + 1);
    float s = 0.f;
    for (int e = lo; e < hi; ++e) s += b0[c * J0 + src1[e]];
    Bs[jj * C1 + c] = s;
}

// ---------- gather-GEMM per node jj: h1T[jj][c][n] = (H0[:,S] @ W0n[S,:])[n,c] + Bs ----------
// grid = 128 blocks (4 per jj) * 128 threads (4 waves); wave = one 16(n) x 32(c) tile
__global__ void k_fgl1_wmma(const float* __restrict__ h0T, const int* __restrict__ src1,
                            const int* __restrict__ dst1, const float* __restrict__ W0n,
                            const float* __restrict__ Bs, float* __restrict__ h1T) {
    int jj    = blockIdx.x >> 2;
    int wave  = threadIdx.x >> 5;
    int lane  = threadIdx.x & 31;
    int l16   = lane & 15;
    int khalf = (lane < 16) ? 0 : 2;
    int n0    = ((blockIdx.x & 3) * 4 + wave) * 16;

    int lo  = lower_bound_i32(dst1, E1CNT, jj);
    int cnt = lower_bound_i32(dst1, E1CNT, jj + 1) - lo;   // uniform per block

    v8f acc0 = {}; v8f acc1 = {};
    for (int k0 = 0; k0 < cnt; k0 += 4) {
        int kA = k0 + khalf;
        v2f a   = {0.f, 0.f};
        v2f bA  = {0.f, 0.f};
        v2f bB  = {0.f, 0.f};
        if (kA < cnt) {
            int s = src1[lo + kA];
            a.x  = h0T[s * N_BATCH + n0 + l16];
            bA.x = W0n[s * C1 + l16];
            bB.x = W0n[s * C1 + 16 + l16];
        }
        if (kA + 1 < cnt) {
            int s = src1[lo + kA + 1];
            a.y  = h0T[s * N_BATCH + n0 + l16];
            bA.y = W0n[s * C1 + l16];
            bB.y = W0n[s * C1 + 16 + l16];
        }
        acc0 = wmma_f32(a, bA, acc0);   // c = 0..15
        acc1 = wmma_f32(a, bB, acc1);   // c = 16..31
    }
    int rowoff = (lane < 16) ? 0 : 8;
#pragma unroll
    for (int v = 0; v < 8; ++v) {
        int n = n0 + rowoff + v;
        h1T[jj * (C1 * N_BATCH) + l16 * N_BATCH + n]        = acc0[v] + Bs[jj * C1 + l16];
        h1T[jj * (C1 * N_BATCH) + (16 + l16) * N_BATCH + n] = acc1[v] + Bs[jj * C1 + 16 + l16];
    }
}

// ---------- per-node projection: z2[n][o*32+jj] = (h1[:, :] @ W1n[jj]^T)[n,o] + b1[o,jj] ----------
// grid = 256 blocks (8 per jj) * 128 threads; wave = one 16(n) x 64(o) tile, K = 32
__global__ void k_fgl1_proj_wmma(const float* __restrict__ h1T, const float* __restrict__ W1n,
                                 const float* __restrict__ b1, float* __restrict__ z2) {
    int jj    = blockIdx.x >> 3;
    int w     = (blockIdx.x & 7) * 4 + (threadIdx.x >> 5);  // 0..31
    int lane  = threadIdx.x & 31;
    int l16   = lane & 15;
    int khalf = (lane < 16) ? 0 : 2;
    int n0    = (w & 15) * 16;
    int o0    = (w >> 4) * 64;

    const float* h1j = h1T + jj * (C1 * N_BATCH);
    const float* W1j = W1n + jj * (C2 * C1);

    v8f acc[4] = {{}, {}, {}, {}};
#pragma unroll
    for (int k0 = 0; k0 < C1; k0 += 4) {
        int k = k0 + khalf;
        v2f a;
        a.x = h1j[k * N_BATCH + n0 + l16];
        a.y = h1j[(k + 1) * N_BATCH + n0 + l16];
#pragma unroll
        for (int t = 0; t < 4; ++t) {
            int o = o0 + t * 16 + l16;
            v2f b;
            b.x = W1j[o * C1 + k];
            b.y = W1j[o * C1 + k + 1];
            acc[t] = wmma_f32(a, b, acc[t]);
        }
    }
    int rowoff = (lane < 16) ? 0 : 8;
#pragma unroll
    for (int t = 0; t < 4; ++t) {
        int o = o0 + t * 16 + l16;
        float bias = b1[o * J1 + jj];
#pragma unroll
        for (int v = 0; v < 8; ++v)
            z2[(n0 + rowoff + v) * (C2 * J1) + o * J1 + jj] = acc[t][v] + bias;
    }
}

// ---------- head GEMM: out(256x768) = z2(256x4096) @ Wl(4096x768) + bl ----------
// grid = 48 blocks (12 p-panels x 4 m-quads) * 128 threads.
// All 4 waves of a block share one 64-wide Wl panel; each 32(K)x64(P) chunk is
// staged into LDS once per block with gfx1250 async global->LDS copies
// (global_load_async_to_lds_b128, ASYNCcnt), then consumed via ds loads.
__global__ void k_final_gemm_wmma(const float* __restrict__ A, const float* __restrict__ Wl,
                                  const float* __restrict__ bl, float* __restrict__ out) {
    const int K  = C2 * J1;  // 4096
    const int KB = 32;       // K-chunk per LDS stage
    __shared__ float tileB[KB * 64];  // 8 KB

    int wave  = threadIdx.x >> 5;
    int lane  = threadIdx.x & 31;
    int l16   = lane & 15;
    int khalf = (lane < 16) ? 0 : 2;
    int pgrp  = blockIdx.x >> 2;       // 0..11
    int mquad = blockIdx.x & 3;        // 0..3
    int p0    = pgrp * 64;
    int m0    = (mquad * 4 + wave) * 16;

    const float* Arow = A + (size_t)(m0 + l16) * K;

    // per-thread async-copy geometry: 4 x b128 = 64B per thread -> 8KB chunk
    int t = threadIdx.x;
    unsigned ldsBase = (unsigned)(uintptr_t)(&tileB[0]);  // low 32 bits = LDS byte offset

    v8f acc[4] = {{}, {}, {}, {}};
    for (int k0 = 0; k0 < K; k0 += KB) {
        const float* sbase = Wl + (size_t)k0 * OUT_DIM + p0;  // uniform SGPR base
#pragma unroll
        for (int r = 0; r < 4; ++r) {
            int f  = t * 4 + r * 512;             // flat float index in tile
            int kk = f >> 6;                      // row in chunk (0..31)
            int pp = f & 63;                      // col in panel (multiple of 4)
            unsigned lds_addr = ldsBase + (unsigned)(f * 4);
            unsigned voff     = (unsigned)((kk * OUT_DIM + pp) * 4);
            asm volatile("global_load_async_to_lds_b128 %0, %1, %2"
                         :: "v"(lds_addr), "v"(voff), "s"(sbase)
                         : "memory");
        }
        asm volatile("s_wait_asynccnt 0x0" ::: "memory");
        __syncthreads();

#pragma unroll
        for (int kc = 0; kc < KB; kc += 4) {
            int k = kc + khalf;
            v2f a = *(const v2f*)(Arow + k0 + k);  // k even -> 8B aligned
#pragma unroll
            for (int tt = 0; tt < 4; ++tt) {
                int pl = tt * 16 + l16;
                v2f b;
                b.x = tileB[k * 64 + pl];
                b.y = tileB[(k + 1) * 64 + pl];
                acc[tt] = wmma_f32(a, b, acc[tt]);
            }
        }
        __syncthreads();
    }

    int rowoff = (lane < 16) ? 0 : 8;
#pragma unroll
    for (int tt = 0; tt < 4; ++tt) {
        int p = p0 + tt * 16 + l16;
        float bias = bl[p];
#pragma unroll
        for (int v = 0; v < 8; ++v)
            out[(m0 + rowoff + v) * OUT_DIM + p] = acc[tt][v] + bias;
    }
}

extern "C" void kernel_launch(void* const* d_in, const int* in_sizes, int n_in,
                              void* d_out, int out_size, void* d_ws, size_t ws_size,
                              hipStream_t stream) {
    const float* x    = (const float*)d_in[0];
    const int*   src0 = (const int*)d_in[1];
    const int*   dst0 = (const int*)d_in[2];
    const int*   src1 = (const int*)d_in[3];
    const int*   dst1 = (const int*)d_in[4];
    const float* V0   = (const float*)d_in[5];
    const float* g0   = (const float*)d_in[6];
    const float* b0   = (const float*)d_in[7];
    const float* V1   = (const float*)d_in[8];
    const float* g1   = (const float*)d_in[9];
    const float* b1   = (const float*)d_in[10];
    const float* Wl   = (const float*)d_in[11];
    const float* bl   = (const float*)d_in[12];
    float* out = (float*)d_out;

    float* ws  = (float*)d_ws;
    float* h0T = ws;                                  // 512*256
    float* W0n = h0T + J0 * N_BATCH;                  // 512*32
    float* W1n = W0n + J0 * C1;                       // 32*128*32
    float* Bs  = W1n + J1 * C2 * C1;                  // 32*32
    float* h1T = Bs + J1 * C1;                        // 32*32*256
    float* z2  = h1T + J1 * C1 * N_BATCH;             // 256*4096
    float* xT  = z2 + N_BATCH * C2 * J1;              // 65536*256 (optional)

    size_t need_full = (size_t)(xT - ws) * sizeof(float)
                     + (size_t)NIN * N_BATCH * sizeof(float);
    int use_xT = (ws_size >= need_full) ? 1 : 0;

    if (use_xT) {
        dim3 tb(32, 8), tg(NIN / 32, N_BATCH / 32);
        k_transpose_x<<<tg, tb, 0, stream>>>(x, xT);
    }
    k_fgl0_gather    <<<J0, N_BATCH, 0, stream>>>(use_xT ? xT : x, src0, dst0, h0T, use_xT);
    k_w0norm         <<<J0, C1,      0, stream>>>(V0, g0, W0n);
    k_w1norm         <<<J1, 256,     0, stream>>>(V1, g1, W1n);
    k_bias0_scatter  <<<J1, C1,      0, stream>>>(src1, dst1, b0, Bs);
    k_fgl1_wmma      <<<J1 * 4, 128, 0, stream>>>(h0T, src1, dst1, W0n, Bs, h1T);
    k_fgl1_proj_wmma <<<J1 * 8, 128, 0, stream>>>(h1T, W1n, b1, z2);
    k_final_gemm_wmma<<<48, 128,     0, stream>>>(z2, Wl, bl, out);
}